// MM_77180562309358
// MI455X (gfx1250) — compile-verified
//
#include <hip/hip_runtime.h>

// Forward-only evaluation of the straight-through VQ reference:
//   h = x_bc . S_c - T_c - 1e-4 ; q = sign(h) ; logits = q @ H ;
//   out_bc = LUT[c, argmax(logits), :]
// Memory-bound (~512MB traffic -> ~22us floor @ 23.3 TB/s). Stage-1 GEMM done
// with V_WMMA_F32_16X16X4_F32 (f32 accumulation matches reference numerics).

#define B_TOT 8192
#define C_CH 64
#define D_DIM 128
#define K_HYP 15
#define M_CODES 16
#define ROWSTRIDE (C_CH * D_DIM) /* 8192 floats */
#define APAD 132                 /* 128 + 4 pad: 16 rows hit distinct LDS banks */
#define ROWS_PER_BLOCK 64
#define WAVES 4

typedef __attribute__((ext_vector_type(2))) float v2f;
typedef __attribute__((ext_vector_type(4))) float v4f;
typedef __attribute__((ext_vector_type(8))) float v8f;

// broadcast lane n of this lane's 16-lane group (wave32 ds_bpermute, byte addr)
__device__ __forceinline__ float bcast16(float v, int n, int lane) {
    int src = (((lane & 16) | n) << 2);
    return __int_as_float(__builtin_amdgcn_ds_bpermute(src, __float_as_int(v)));
}
__device__ __forceinline__ float xshf_f(float v, int off, int lane) {
    return __int_as_float(__builtin_amdgcn_ds_bpermute((lane ^ off) << 2, __float_as_int(v)));
}
__device__ __forceinline__ int xshf_i(int v, int off, int lane) {
    return __builtin_amdgcn_ds_bpermute((lane ^ off) << 2, v);
}

__global__ __launch_bounds__(WAVES * 32)
void vq_fwd_kernel(const float* __restrict__ x, const float* __restrict__ S,
                   const float* __restrict__ H, const float* __restrict__ T,
                   const float* __restrict__ LUT, float* __restrict__ out) {
    __shared__ float sS[D_DIM * 16];            // S[c] padded to 16 hyps (8 KB)
    __shared__ float sA[ROWS_PER_BLOCK * APAD]; // x tile, padded rows (33 KB)

    const int tid  = threadIdx.x;
    const int c    = blockIdx.y;
    const int b0   = blockIdx.x * ROWS_PER_BLOCK;
    const int lane = tid & 31;
    const int wave = tid >> 5;
    const int half = (lane >> 4) & 1; // 0: lanes 0-15, 1: lanes 16-31
    const int ln   = lane & 15;

    // per-lane constants (lane ln owns hyp/code column ln)
    const float tval = (ln < K_HYP) ? T[c * K_HYP + ln] : 0.0f;
    float Hc[K_HYP];
#pragma unroll
    for (int n = 0; n < K_HYP; ++n) Hc[n] = H[n * M_CODES + ln];

    // warm the channel's LUT slab (global_prefetch_b8)
    __builtin_prefetch(&LUT[(size_t)c * M_CODES * D_DIM], 0, 0);

    // stage S[c] (128 x 15 -> 128 x 16, col 15 zero) into LDS
    for (int i = tid; i < D_DIM * 16; i += WAVES * 32) {
        const int k = i >> 4, n = i & 15;
        sS[i] = (n < K_HYP) ? S[((size_t)c * D_DIM + k) * K_HYP + n] : 0.0f;
    }
    // stage 64 x 128 f32 x-tile, fully coalesced b128 loads (one row / iter / wave)
    {
        const int chunk = tid & 31; // 4-float chunk within a row
        const int rsub  = tid >> 5;
#pragma unroll 4
        for (int it = 0; it < 16; ++it) {
            const int row = it * 4 + rsub;
            const v4f val = *(const v4f*)(x + (size_t)(b0 + row) * ROWSTRIDE
                                            + c * D_DIM + chunk * 4);
            *(v4f*)(&sA[row * APAD + chunk * 4]) = val;
        }
    }
    __syncthreads();

    // ---- stage 1: h = A(16x128) @ S_c(128x16) via 32 chained f32 WMMAs ----
    // A-frag (16x4 f32): lanes 0-15 M=0..15 {K0,K1}; lanes 16-31 {K2,K3}
    // B-frag (4x16 f32): lane n holds {S[K0..1][n]} / {S[K2..3][n]}
    const int arow = wave * 16 + ln;
    v8f acc = {};
#pragma unroll
    for (int kb = 0; kb < D_DIM; kb += 4) {
        const int ka = kb + 2 * half;
        v2f a, b;
        a.x = sA[arow * APAD + ka];
        a.y = sA[arow * APAD + ka + 1];
        b.x = sS[ka * 16 + ln];
        b.y = sS[(ka + 1) * 16 + ln];
        acc = __builtin_amdgcn_wmma_f32_16x16x4_f32(
            /*neg_a=*/false, a, /*neg_b=*/false, b,
            /*c_mod=*/(short)0, acc, /*reuse_a=*/false, /*reuse_b=*/false);
    }

    // q = sign(h); D-layout: VGPR j = row (j + 8*half), lane column n = ln
    float q[8];
#pragma unroll
    for (int j = 0; j < 8; ++j) {
        const float h = acc[j] - tval - 1e-4f;
        q[j] = (h > 0.0f) ? 1.0f : ((h < 0.0f) ? -1.0f : 0.0f);
    }

#pragma unroll
    for (int j = 0; j < 8; ++j) {
        // logits[m=ln] = sum_n q[n] * H[n][m] via 16-lane broadcasts
        float s = 0.0f;
#pragma unroll
        for (int n = 0; n < K_HYP; ++n)
            s = fmaf(bcast16(q[j], n, lane), Hc[n], s);
        // argmax over m across the 16-lane group (ties -> lowest index)
        float bv = s;
        int   bi = ln;
#pragma unroll
        for (int off = 8; off > 0; off >>= 1) {
            const float ov = xshf_f(bv, off, lane);
            const int   oi = xshf_i(bi, off, lane);
            const bool take = (ov > bv) || (ov == bv && oi < bi);
            bv = take ? ov : bv;
            bi = take ? oi : bi;
        }
        // gather LUT[c, m*, :] -> out row (16 lanes x 8 floats, b128 each)
        const float* lr = LUT + (((size_t)c * M_CODES) + bi) * D_DIM;
        float* orow = out + (size_t)(b0 + wave * 16 + half * 8 + j) * ROWSTRIDE
                          + c * D_DIM;
        const v4f u0 = *(const v4f*)(lr + ln * 8);
        const v4f u1 = *(const v4f*)(lr + ln * 8 + 4);
        *(v4f*)(orow + ln * 8)     = u0;
        *(v4f*)(orow + ln * 8 + 4) = u1;
    }
}

extern "C" void kernel_launch(void* const* d_in, const int* in_sizes, int n_in,
                              void* d_out, int out_size, void* d_ws, size_t ws_size,
                              hipStream_t stream) {
    const float* x   = (const float*)d_in[0];
    const float* S   = (const float*)d_in[1];
    const float* H   = (const float*)d_in[2];
    const float* T   = (const float*)d_in[3];
    const float* LUT = (const float*)d_in[4];
    float* out = (float*)d_out;

    dim3 grid(B_TOT / ROWS_PER_BLOCK, C_CH); // 128 x 64 blocks
    dim3 block(WAVES * 32);                  // 4 wave32
    vq_fwd_kernel<<<grid, block, 0, stream>>>(x, S, H, T, LUT, out);
}